// SGEncoder_22376779612491
// MI455X (gfx1250) — compile-verified
//
#include <hip/hip_runtime.h>

#define DD 300      // feature dim
#define DP 308      // LDS row stride (dwords): multiple of 4 (b128 LDS ops), conflict-free A reads
#define NP 320      // padded weight column count (20 tiles of 16)
#define GG 128      // number of graphs
#define TT 4        // tokens per node/edge
#define QQ 75       // DD/4 float4s per feature row
#define WROWS 3000  // total concatenated weight rows (all six matrices)
#define NPACK ((WROWS / 4) * 2 * NP)  // packed float2 count

typedef float v2f __attribute__((ext_vector_type(2)));
typedef float v8f __attribute__((ext_vector_type(8)));

__device__ __forceinline__ v8f zero8() {
  v8f z = {0.f, 0.f, 0.f, 0.f, 0.f, 0.f, 0.f, 0.f};
  return z;
}

// D = A*B + C for fp32 16x16x4 (exact fp32 — workload is HBM-bound, no need for bf16)
__device__ __forceinline__ v8f wmma4(v2f a, v2f b, v8f c) {
  return __builtin_amdgcn_wmma_f32_16x16x4_f32(false, a, false, b, (short)0, c,
                                               false, false);
}

// A fragment 16x4 f32 from LDS: lane m=l&15; K=k (+2 for hi half lanes)
__device__ __forceinline__ v2f ldsA(const float* buf, int k) {
  int l = threadIdx.x & 31;
  int m = l & 15;
  int kk = k + ((l >> 4) << 1);
  v2f a;
  a.x = buf[m * DP + kk];
  a.y = buf[m * DP + kk + 1];
  return a;
}

// B fragment 4x16 f32 from fragment-packed weights: one b64 load per lane.
// Packed layout: Bp[(k/4)*2 + half][n] = {W[k/4*4 + half*2][n], W[...+1][n]}
__device__ __forceinline__ v2f gldB(const float2* __restrict__ Bp, int krow,
                                    int n0) {
  int l = threadIdx.x & 31;
  float2 b2 =
      Bp[((size_t)((krow >> 2) * 2 + (l >> 4))) * NP + n0 + (l & 15)];
  v2f b;
  b.x = b2.x;
  b.y = b2.y;
  return b;
}

// One K=300 segment of the 16x320 GEMM. Each wave owns 5 contiguous column
// tiles (wave*5 .. wave*5+4): A fragment reused 5x, 5 independent WMMA chains.
#define GEMM_SEG(ACC, BUF, WP, ROW0)                                          \
  _Pragma("unroll 2") for (int k_ = 0; k_ < DD; k_ += 4) {                    \
    v2f a_ = ldsA(BUF, k_);                                                   \
    _Pragma("unroll") for (int it_ = 0; it_ < 5; ++it_) {                     \
      ACC[it_] =                                                              \
          wmma4(a_, gldB(WP, (ROW0) + k_, (wave * 5 + it_) * 16), ACC[it_]);  \
    }                                                                         \
  }

#define ZERO_ACC(ACC) \
  _Pragma("unroll") for (int i_ = 0; i_ < 5; ++i_) ACC[i_] = zero8();

// bias + ReLU -> bufH  (C layout: row = j + 8*(lane>=16), col = lane&15)
#define EPI_RELU_H(ACC, BIAS)                                                 \
  {                                                                           \
    int lane_ = threadIdx.x & 31;                                             \
    int nl_ = lane_ & 15;                                                     \
    int mh_ = (lane_ >> 4) << 3;                                              \
    _Pragma("unroll") for (int it_ = 0; it_ < 5; ++it_) {                     \
      int n_ = (wave * 5 + it_) * 16 + nl_;                                   \
      if (n_ < DD) {                                                          \
        float bv_ = (BIAS)[n_];                                               \
        _Pragma("unroll") for (int j_ = 0; j_ < 8; ++j_) {                    \
          float v_ = ACC[it_][j_] + bv_;                                      \
          bufH[(mh_ + j_) * DP + n_] = v_ > 0.f ? v_ : 0.f;                   \
        }                                                                     \
      }                                                                       \
    }                                                                         \
  }

// ---------------- small prep kernels ----------------

// Pack the six [K,300] weight matrices (rows concatenated: ew1:0-899,
// ew2:900-1199, n1w1:1200-1799, n1w2:1800-2099, n2w1:2100-2699,
// n2w2:2700-2999) into WMMA-fragment order with zero-padded columns:
// bp[(kg*2+half)*320 + n] = {W[kg*4+half*2][n], W[kg*4+half*2+1][n]}
__global__ void k_pack_w(const float* __restrict__ ew1, const float* __restrict__ ew2,
                         const float* __restrict__ n1w1, const float* __restrict__ n1w2,
                         const float* __restrict__ n2w1, const float* __restrict__ n2w2,
                         float2* __restrict__ bp) {
  int idx = blockIdx.x * blockDim.x + threadIdx.x;
  if (idx >= NPACK) return;
  int n = idx % NP;
  int g = idx / NP;          // kg*2 + half
  int r0 = (g >> 1) * 4 + (g & 1) * 2;  // first of the row pair, 0..2998
  float2 v = make_float2(0.f, 0.f);
  if (n < DD) {
    const float* src;
    int rl;
    if (r0 < 900) { src = ew1; rl = r0; }
    else if (r0 < 1200) { src = ew2; rl = r0 - 900; }
    else if (r0 < 1800) { src = n1w1; rl = r0 - 1200; }
    else if (r0 < 2100) { src = n1w2; rl = r0 - 1800; }
    else if (r0 < 2700) { src = n2w1; rl = r0 - 2100; }
    else { src = n2w2; rl = r0 - 2700; }
    v.x = src[(size_t)rl * DD + n];
    v.y = src[(size_t)(rl + 1) * DD + n];
  }
  bp[idx] = v;
}

__global__ void k_embed_x(const float* __restrict__ emb,
                          const int* __restrict__ tok, float* __restrict__ x,
                          int N) {
  size_t idx = (size_t)blockIdx.x * blockDim.x + threadIdx.x;
  if (idx >= (size_t)N * QQ) return;
  int n = (int)(idx / QQ);
  int q = (int)(idx % QQ);
  float4 s = make_float4(0.f, 0.f, 0.f, 0.f);
#pragma unroll
  for (int u = 0; u < TT; ++u) {
    float4 e = ((const float4*)(emb + (size_t)tok[n * TT + u] * DD))[q];
    s.x += e.x; s.y += e.y; s.z += e.z; s.w += e.w;
  }
  ((float4*)(x + (size_t)n * DD))[q] = s;
}

__global__ void k_fill_sign(float* __restrict__ sign, int E) {
  int e = blockIdx.x * blockDim.x + threadIdx.x;
  if (e < E) sign[e] = 1.f;
}

__global__ void k_scatter_sign(const int* __restrict__ added,
                               float* __restrict__ sign, int M) {
  int i = blockIdx.x * blockDim.x + threadIdx.x;
  if (i < M) sign[added[i]] = -1.f;
}

__global__ void k_edge_cnt(const int* __restrict__ eidx, float* __restrict__ cnt,
                           int E) {
  int e = blockIdx.x * blockDim.x + threadIdx.x;
  if (e < E) atomicAdd(&cnt[eidx[E + e]], 1.f);
}

__global__ void k_node_cnt(const int* __restrict__ batch,
                           float* __restrict__ gcnt, int N) {
  int n = blockIdx.x * blockDim.x + threadIdx.x;
  if (n < N) atomicAdd(&gcnt[batch[n]], 1.f);
}

// ---------------- fused edge pipeline ----------------
// Per block: 16 edges. Builds [x_row | x_col | ea] segment-wise in LDS,
// runs edge MLP (K=900), writes edge_enc to d_out, then message MLP (K=600),
// atomically scatters m into aggsum[col].

__global__ __launch_bounds__(128) void k_edge_fused(
    const float* __restrict__ x, const float* __restrict__ emb,
    const int* __restrict__ etok, const float* __restrict__ sign,
    const int* __restrict__ eidx, const float2* __restrict__ ew1p,
    const float* __restrict__ eb1, const float2* __restrict__ ew2p,
    const float* __restrict__ eb2, const float2* __restrict__ n1w1p,
    const float* __restrict__ n1b1, const float2* __restrict__ n1w2p,
    const float* __restrict__ n1b2, float* __restrict__ edge_out,
    float* __restrict__ aggsum, int E) {
  __shared__ float bufX[16 * DP];  // x[row], reused by both MLPs
  __shared__ float bufS[16 * DP];  // x[col] -> ea -> edge_enc
  __shared__ float bufH[16 * DP];  // ReLU hidden
  __shared__ int rowi[16], coli[16];
  __shared__ float sgn[16];

  const int e0 = blockIdx.x * 16;
  const int tid = threadIdx.x;
  const int wave = tid >> 5;

  if (tid < 16) {
    rowi[tid] = eidx[e0 + tid];
    coli[tid] = eidx[E + e0 + tid];
    sgn[tid] = sign[e0 + tid];
  }
  __syncthreads();

  for (int t = tid; t < 16 * QQ; t += 128) {
    int i = t / QQ, q = t - i * QQ;
    *(float4*)&bufX[i * DP + q * 4] =
        ((const float4*)(x + (size_t)rowi[i] * DD))[q];
    *(float4*)&bufS[i * DP + q * 4] =
        ((const float4*)(x + (size_t)coli[i] * DD))[q];
  }
  __syncthreads();

  // ---- edge MLP layer 1: h = ReLU([x_row|x_col|ea] @ ew1 + eb1)
  v8f acc[5];
  ZERO_ACC(acc);
  GEMM_SEG(acc, bufX, ew1p, 0);    // K rows 0..299   (x_row)
  GEMM_SEG(acc, bufS, ew1p, 300);  // K rows 300..599 (x_col)
  __syncthreads();
  // restage bufS with ea = sign * sum_t emb[edge_attr_tok]
  for (int t = tid; t < 16 * QQ; t += 128) {
    int i = t / QQ, q = t - i * QQ;
    int e = e0 + i;
    float4 s = make_float4(0.f, 0.f, 0.f, 0.f);
#pragma unroll
    for (int u = 0; u < TT; ++u) {
      float4 v = ((const float4*)(emb + (size_t)etok[e * TT + u] * DD))[q];
      s.x += v.x; s.y += v.y; s.z += v.z; s.w += v.w;
    }
    float g = sgn[i];
    s.x *= g; s.y *= g; s.z *= g; s.w *= g;
    *(float4*)&bufS[i * DP + q * 4] = s;
  }
  __syncthreads();
  GEMM_SEG(acc, bufS, ew1p, 600);  // K rows 600..899 (ea)
  EPI_RELU_H(acc, eb1);
  __syncthreads();

  // ---- edge MLP layer 2: edge_enc = h @ ew2 + eb2
  v8f acc2[5];
  ZERO_ACC(acc2);
  GEMM_SEG(acc2, bufH, ew2p, 0);
  {
    int lane = tid & 31, nl = lane & 15, mh = (lane >> 4) << 3;
#pragma unroll
    for (int it = 0; it < 5; ++it) {
      int n = (wave * 5 + it) * 16 + nl;
      if (n < DD) {
        float bv = eb2[n];
#pragma unroll
        for (int j = 0; j < 8; ++j) {
          float v = acc2[it][j] + bv;
          bufS[(mh + j) * DP + n] = v;                   // feed msg MLP
          edge_out[(size_t)(e0 + mh + j) * DD + n] = v;  // output 2
        }
      }
    }
  }
  __syncthreads();

  // ---- message MLP layer 1: hm = ReLU([x_row|edge_enc] @ n1w1 + n1b1)
  v8f acc3[5];
  ZERO_ACC(acc3);
  GEMM_SEG(acc3, bufX, n1w1p, 0);
  GEMM_SEG(acc3, bufS, n1w1p, 300);
  EPI_RELU_H(acc3, n1b1);
  __syncthreads();

  // ---- message MLP layer 2 + scatter-add into aggsum[col]
  v8f acc4[5];
  ZERO_ACC(acc4);
  GEMM_SEG(acc4, bufH, n1w2p, 0);
  {
    int lane = tid & 31, nl = lane & 15, mh = (lane >> 4) << 3;
#pragma unroll
    for (int it = 0; it < 5; ++it) {
      int n = (wave * 5 + it) * 16 + nl;
      if (n < DD) {
        float bv = n1b2[n];
#pragma unroll
        for (int j = 0; j < 8; ++j) {
          float v = acc4[it][j] + bv;
          atomicAdd(&aggsum[(size_t)coli[mh + j] * DD + n], v);
        }
      }
    }
  }
}

// ---------------- fused node pipeline ----------------

__global__ __launch_bounds__(128) void k_node_fused(
    const float* __restrict__ x, const float* __restrict__ aggsum,
    const float* __restrict__ cnt, const int* __restrict__ batch,
    const float2* __restrict__ n2w1p, const float* __restrict__ n2b1,
    const float2* __restrict__ n2w2p, const float* __restrict__ n2b2,
    float* __restrict__ xenc, float* __restrict__ gsum,
    float* __restrict__ gss, int N) {
  __shared__ float bufX[16 * DP];
  __shared__ float bufS[16 * DP];
  __shared__ float bufH[16 * DP];
  __shared__ float rcnt[16];

  const int n0 = blockIdx.x * 16;
  const int tid = threadIdx.x;
  const int wave = tid >> 5;

  if (tid < 16) rcnt[tid] = 1.f / fmaxf(cnt[n0 + tid], 1.f);
  __syncthreads();
  for (int t = tid; t < 16 * QQ; t += 128) {
    int i = t / QQ, q = t - i * QQ;
    *(float4*)&bufX[i * DP + q * 4] =
        ((const float4*)(x + (size_t)(n0 + i) * DD))[q];
    float4 a = ((const float4*)(aggsum + (size_t)(n0 + i) * DD))[q];
    float r = rcnt[i];
    a.x *= r; a.y *= r; a.z *= r; a.w *= r;
    *(float4*)&bufS[i * DP + q * 4] = a;
  }
  __syncthreads();

  v8f acc[5];
  ZERO_ACC(acc);
  GEMM_SEG(acc, bufX, n2w1p, 0);
  GEMM_SEG(acc, bufS, n2w1p, 300);
  EPI_RELU_H(acc, n2b1);
  __syncthreads();

  v8f acc2[5];
  ZERO_ACC(acc2);
  GEMM_SEG(acc2, bufH, n2w2p, 0);
  {
    int lane = tid & 31, nl = lane & 15, mh = (lane >> 4) << 3;
#pragma unroll
    for (int it = 0; it < 5; ++it) {
      int n = (wave * 5 + it) * 16 + nl;
      if (n < DD) {
        float bv = n2b2[n];
#pragma unroll
        for (int j = 0; j < 8; ++j) {
          float v = acc2[it][j] + bv;
          bufS[(mh + j) * DP + n] = v;
          xenc[(size_t)(n0 + mh + j) * DD + n] = v;
        }
      }
    }
  }
  __syncthreads();
  // per-graph layernorm partials
  if (tid < 16) {
    float s = 0.f, ss = 0.f;
    for (int d = 0; d < DD; ++d) {
      float v = bufS[tid * DP + d];
      s += v;
      ss += v * v;
    }
    int b = batch[n0 + tid];
    atomicAdd(&gsum[b], s);
    atomicAdd(&gss[b], ss);
  }
}

// ---------------- layernorm finalize ----------------

__global__ void k_gstats(const float* __restrict__ gcnt,
                         const float* __restrict__ gsum,
                         const float* __restrict__ gss,
                         float* __restrict__ gmean, float* __restrict__ ginv) {
  int g = threadIdx.x;
  if (g < GG) {
    float denom = fmaxf(gcnt[g] * (float)DD, 1.f);
    float mean = gsum[g] / denom;
    float var = gss[g] / denom - mean * mean;
    gmean[g] = mean;
    ginv[g] = rsqrtf(var + 1e-5f);
  }
}

__global__ void k_norm(const float* __restrict__ xenc,
                       const int* __restrict__ batch,
                       const float* __restrict__ gmean,
                       const float* __restrict__ ginv,
                       const float* __restrict__ lw,
                       const float* __restrict__ lb, float* __restrict__ out,
                       int N) {
  size_t idx = (size_t)blockIdx.x * blockDim.x + threadIdx.x;
  if (idx >= (size_t)N * QQ) return;
  int n = (int)(idx / QQ);
  int q = (int)(idx % QQ);
  int b = batch[n];
  float m = gmean[b], iv = ginv[b];
  float4 v = ((const float4*)(xenc + (size_t)n * DD))[q];
  float4 w = ((const float4*)lw)[q];
  float4 c = ((const float4*)lb)[q];
  float4 o;
  o.x = (v.x - m) * iv * w.x + c.x;
  o.y = (v.y - m) * iv * w.y + c.y;
  o.z = (v.z - m) * iv * w.z + c.z;
  o.w = (v.w - m) * iv * w.w + c.w;
  ((float4*)(out + (size_t)n * DD))[q] = o;
}

extern "C" void kernel_launch(void* const* d_in, const int* in_sizes, int n_in,
                              void* d_out, int out_size, void* d_ws,
                              size_t ws_size, hipStream_t stream) {
  const int* x_tok = (const int*)d_in[0];
  const int* etok = (const int*)d_in[1];
  const int* eidx = (const int*)d_in[2];
  const int* added = (const int*)d_in[3];
  const int* batch = (const int*)d_in[4];
  const float* emb = (const float*)d_in[5];
  const float* ew1 = (const float*)d_in[6];
  const float* eb1 = (const float*)d_in[7];
  const float* ew2 = (const float*)d_in[8];
  const float* eb2 = (const float*)d_in[9];
  const float* n1w1 = (const float*)d_in[10];
  const float* n1b1 = (const float*)d_in[11];
  const float* n1w2 = (const float*)d_in[12];
  const float* n1b2 = (const float*)d_in[13];
  const float* n2w1 = (const float*)d_in[14];
  const float* n2b1 = (const float*)d_in[15];
  const float* n2w2 = (const float*)d_in[16];
  const float* n2b2 = (const float*)d_in[17];
  const float* lnw = (const float*)d_in[18];
  const float* lnb = (const float*)d_in[19];

  const int N = in_sizes[0] / TT;
  const int E = in_sizes[1] / TT;
  const int M = in_sizes[3];

  // workspace carve-up (~367 MB)
  float* ws = (float*)d_ws;
  float* x = ws;                          // N*300
  float* aggsum = x + (size_t)N * DD;     // N*300
  float* xenc = aggsum + (size_t)N * DD;  // N*300
  float* sign = xenc + (size_t)N * DD;    // E
  float* cnt = sign + E;                  // N
  float* gcnt = cnt + N;                  // GG (contiguous with gsum, gss)
  float* gsum = gcnt + GG;
  float* gss = gsum + GG;
  float* gmean = gss + GG;
  float* ginv = gmean + GG;
  float2* bp = (float2*)(ginv + GG);      // NPACK float2 packed weights

  // packed base (float2 units) for matrix starting at concat row R0:
  // (R0/4)*2*NP
  const float2* ew1p = bp;
  const float2* ew2p = bp + (size_t)(900 / 4) * 2 * NP;
  const float2* n1w1p = bp + (size_t)(1200 / 4) * 2 * NP;
  const float2* n1w2p = bp + (size_t)(1800 / 4) * 2 * NP;
  const float2* n2w1p = bp + (size_t)(2100 / 4) * 2 * NP;
  const float2* n2w2p = bp + (size_t)(2700 / 4) * 2 * NP;

  float* xout = (float*)d_out;
  float* eout = xout + (size_t)N * DD;

  hipMemsetAsync(aggsum, 0, (size_t)N * DD * sizeof(float), stream);
  hipMemsetAsync(cnt, 0, (size_t)N * sizeof(float), stream);
  hipMemsetAsync(gcnt, 0, (size_t)3 * GG * sizeof(float), stream);

  const int thr = 256;
  const size_t nq = (size_t)N * QQ;

  k_pack_w<<<(NPACK + thr - 1) / thr, thr, 0, stream>>>(ew1, ew2, n1w1, n1w2,
                                                        n2w1, n2w2, bp);
  k_embed_x<<<(unsigned)((nq + thr - 1) / thr), thr, 0, stream>>>(emb, x_tok, x, N);
  k_fill_sign<<<(E + thr - 1) / thr, thr, 0, stream>>>(sign, E);
  k_scatter_sign<<<(M + thr - 1) / thr, thr, 0, stream>>>(added, sign, M);
  k_edge_cnt<<<(E + thr - 1) / thr, thr, 0, stream>>>(eidx, cnt, E);
  k_node_cnt<<<(N + thr - 1) / thr, thr, 0, stream>>>(batch, gcnt, N);

  k_edge_fused<<<E / 16, 128, 0, stream>>>(x, emb, etok, sign, eidx, ew1p, eb1,
                                           ew2p, eb2, n1w1p, n1b1, n1w2p, n1b2,
                                           eout, aggsum, E);

  k_node_fused<<<N / 16, 128, 0, stream>>>(x, aggsum, cnt, batch, n2w1p, n2b1,
                                           n2w2p, n2b2, xenc, gsum, gss, N);

  k_gstats<<<1, GG, 0, stream>>>(gcnt, gsum, gss, gmean, ginv);

  k_norm<<<(unsigned)((nq + thr - 1) / thr), thr, 0, stream>>>(
      xenc, batch, gmean, ginv, lnw, lnb, xout, N);
}